// SRVQC_53369263620426
// MI455X (gfx1250) — compile-verified
//
#include <hip/hip_runtime.h>

#define NQ      12
#define KSTEPS  6
#define BATCH   256
#define SEQLEN  128
#define DEMB    512
#define NCLS    8
#define QDIM    4096          // 2^NQ
#define PI_F    3.14159274f   // float32(np.pi)

typedef float    v2f  __attribute__((ext_vector_type(2)));
typedef float    v8f  __attribute__((ext_vector_type(8)));
typedef int      v4i  __attribute__((ext_vector_type(4)));

#if __has_builtin(__builtin_amdgcn_global_load_async_to_lds_b128) && \
    __has_builtin(__builtin_amdgcn_s_wait_asynccnt)
#define HAVE_ASYNC_LDS 1
typedef __attribute__((address_space(1))) v4i* gas_v4i;   // global int4*
typedef __attribute__((address_space(3))) v4i* las_v4i;   // LDS int4*
#else
#define HAVE_ASYNC_LDS 0
#endif

// ---------------------------------------------------------------------------
// Kernel 1: embedding gather + masked mean pool (into LDS) + WMMA projection
//   grid = 16 blocks, 256 threads; block handles samples [16*blk, 16*blk+16)
//   writes x[256][12] (post tanh * pi) into d_ws
// ---------------------------------------------------------------------------
__global__ __launch_bounds__(256) void encode_kernel(
    const int* __restrict__ ids,     // [256][128]
    const int* __restrict__ amask,   // [256][128]
    const float* __restrict__ emb,   // [32000][512]
    const float* __restrict__ pw,    // [12][512]
    const float* __restrict__ pb,    // [12]
    float* __restrict__ xout)        // [256][12]
{
    __shared__ float pooled_s[16][DEMB];   // 32 KB  (A tile)
    __shared__ float pw_s[16][DEMB];       // 32 KB  (B tile, rows 12..15 = 0)
    const int tid = threadIdx.x;
    const int b0  = blockIdx.x * 16;

    // ---- stage zero-padded proj_w into LDS (unpredicated B loads later) ----
#if HAVE_ASYNC_LDS
    // Async memory->LDS copy of the 12x512 f32 proj_w (6144 dwords = 1536 x b128),
    // issued before the HBM-bound gather so it completes for free in its shadow.
#pragma unroll
    for (int it = 0; it < (NQ * DEMB / 4) / 256; ++it) {   // 6 x b128 per thread
        const int idx4 = (tid + it * 256) * 4;             // dword index, 16B aligned
        __builtin_amdgcn_global_load_async_to_lds_b128(
            (gas_v4i)(pw + idx4), (las_v4i)(&pw_s[0][0] + idx4), 0, 0);
    }
    // zero padding rows 12..15
#pragma unroll
    for (int it = 0; it < (4 * DEMB) / 256; ++it) {        // 8 per thread
        const int idx = tid + it * 256;
        pw_s[NQ + (idx >> 9)][idx & (DEMB - 1)] = 0.f;
    }
#else
    for (int idx = tid; idx < 16 * DEMB; idx += 256) {
        const int r  = idx >> 9;          // /512
        const int cc = idx & (DEMB - 1);
        pw_s[r][cc] = (r < NQ) ? pw[r * DEMB + cc] : 0.f;
    }
#endif

    // ---- phase 1: gather + pool, 2 dims per thread per sample ----
    for (int ss = 0; ss < 16; ++ss) {
        const int b = b0 + ss;
        float a0 = 0.f, a1 = 0.f, ms = 0.f;
        int id = ids[b * SEQLEN];
        for (int s = 0; s < SEQLEN; ++s) {
            const int id_next = (s + 1 < SEQLEN) ? ids[b * SEQLEN + s + 1] : id;
            __builtin_prefetch(emb + (size_t)id_next * DEMB + tid, 0, 1);
            const float m = (float)amask[b * SEQLEN + s];
            const float* row = emb + (size_t)id * DEMB;
            a0 += m * row[tid];
            a1 += m * row[tid + 256];
            ms += m;
            id = id_next;
        }
        const float inv = 1.0f / fmaxf(ms, 1.0f);
        pooled_s[ss][tid]       = a0 * inv;
        pooled_s[ss][tid + 256] = a1 * inv;
    }
#if HAVE_ASYNC_LDS
    __builtin_amdgcn_s_wait_asynccnt(0);
#endif
    __syncthreads();

    // ---- phase 2: 16x16 WMMA tile: pooled[16x512] x proj_w^T[512x16] ----
    // All 8 waves compute the identical tile (EXEC all-ones everywhere);
    // only wave 0 stores.
    const int lane = tid & 31;
    const int half = lane >> 4;    // 0: lanes 0-15, 1: lanes 16-31
    const int l16  = lane & 15;

    v8f c = {0.f, 0.f, 0.f, 0.f, 0.f, 0.f, 0.f, 0.f};

    // V_WMMA_F32_16X16X4_F32:
    //   A 16x4: lane(l16) = M row; VGPR0/1 hold K = (2*half)+0 / +1
    //   B  4x16: lane(l16) = N col; same K striping (symmetric layout)
#pragma unroll 4
    for (int k0 = 0; k0 < DEMB; k0 += 4) {
        const int ka = k0 + 2 * half;                       // even -> 8B aligned
        const v2f a  = *(const v2f*)&pooled_s[l16][ka];     // ds_load_b64
        const v2f bm = *(const v2f*)&pw_s[l16][ka];         // ds_load_b64
        c = __builtin_amdgcn_wmma_f32_16x16x4_f32(
                false, a, false, bm, (short)0, c, false, false);
    }

    if (tid < 32 && l16 < NQ) {
        // C/D layout: VGPR j, lanes 0-15 -> M=j, lanes 16-31 -> M=j+8; N=l16
        const float bias = pb[l16];
#pragma unroll
        for (int j = 0; j < 8; ++j) {
            const int row = b0 + j + 8 * half;
            xout[row * NQ + l16] = tanhf(c[j] + bias) * PI_F;
        }
    }
}

// ---------------------------------------------------------------------------
// Kernel 2: state-vector simulation, one workgroup (256 thr) per sample.
// State lives in LDS (re/im, 32 KB).  Writes readouts[K][B][NQ].
// ---------------------------------------------------------------------------
__device__ __forceinline__ void apply1q_g(
    float* re, float* im, int q,
    float u00r, float u00i, float u01r, float u01i,
    float u10r, float u10i, float u11r, float u11i, int tid)
{
    const int sh    = NQ - 1 - q;   // log2(right)
    const int right = 1 << sh;
#pragma unroll
    for (int it = 0; it < QDIM / 2 / 256; ++it) {
        const int p   = tid + it * 256;
        const int low = p & (right - 1);
        const int i0  = ((p >> sh) << (sh + 1)) | low;
        const int i1  = i0 | right;
        const float ar = re[i0], ai = im[i0];
        const float br = re[i1], bi = im[i1];
        re[i0] = u00r * ar - u00i * ai + u01r * br - u01i * bi;
        im[i0] = u00r * ai + u00i * ar + u01r * bi + u01i * br;
        re[i1] = u10r * ar - u10i * ai + u11r * br - u11i * bi;
        im[i1] = u10r * ai + u10i * ar + u11r * bi + u11i * br;
    }
}

__device__ __forceinline__ void cnot_g(float* re, float* im, int cw, int tw, int tid)
{
    const int cb = NQ - 1 - cw, tb = NQ - 1 - tw;
    const int lo = (cb < tb) ? cb : tb;
    const int hi = (cb < tb) ? tb : cb;
#pragma unroll
    for (int it = 0; it < QDIM / 4 / 256; ++it) {
        const int p = tid + it * 256;
        int i = p;
        i = ((i >> lo) << (lo + 1)) | (i & ((1 << lo) - 1));
        i = ((i >> hi) << (hi + 1)) | (i & ((1 << hi) - 1));
        i |= (1 << cb);              // control=1, target bit currently 0
        const int j = i | (1 << tb);
        const float tr = re[i], ti = im[i];
        re[i] = re[j]; im[i] = im[j];
        re[j] = tr;    im[j] = ti;
    }
}

__global__ __launch_bounds__(256) void qsim_kernel(
    const float* __restrict__ x,      // [256][12]
    const float* __restrict__ theta,  // [6][12][3]
    float* __restrict__ ro)           // [6][256][12]
{
    __shared__ float s_re[QDIM];
    __shared__ float s_im[QDIM];
    __shared__ float s_acc[NQ];

    const int b   = blockIdx.x;
    const int tid = threadIdx.x;

#pragma unroll
    for (int it = 0; it < QDIM / 256; ++it) {
        s_re[tid + it * 256] = 0.f;
        s_im[tid + it * 256] = 0.f;
    }
    if (tid == 0) s_re[0] = 1.0f;

    float xv[NQ];
#pragma unroll
    for (int i = 0; i < NQ; ++i) xv[i] = x[b * NQ + i];
    __syncthreads();

    for (int k = 0; k < KSTEPS; ++k) {
        // data re-upload: RY(x_i) on wire i
        for (int i = 0; i < NQ; ++i) {
            float s, cgt;
            sincosf(0.5f * xv[i], &s, &cgt);
            apply1q_g(s_re, s_im, i, cgt, 0.f, -s, 0.f, s, 0.f, cgt, 0.f, tid);
            __syncthreads();
        }
        // trainable Rot(phi, theta, omega) = RZ(om) RY(th) RZ(phi)
        for (int i = 0; i < NQ; ++i) {
            const float ph = theta[(k * NQ + i) * 3 + 0];
            const float th = theta[(k * NQ + i) * 3 + 1];
            const float om = theta[(k * NQ + i) * 3 + 2];
            float s2, c2;
            sincosf(0.5f * th, &s2, &c2);
            const float apo = 0.5f * (ph + om), amo = 0.5f * (ph - om);
            const float epr = cosf(apo), epi = -sinf(apo);   // exp(-i(ph+om)/2)
            const float emr = cosf(amo), emi =  sinf(amo);   // exp(+i(ph-om)/2)
            apply1q_g(s_re, s_im, i,
                      epr * c2,  epi * c2,      // u00 = ep*c
                     -emr * s2, -emi * s2,      // u01 = -em*s
                      emr * s2, -emi * s2,      // u10 = conj(em)*s
                      epr * c2, -epi * c2,      // u11 = conj(ep)*c
                      tid);
            __syncthreads();
        }
        // ring of CNOTs
        for (int i = 0; i < NQ - 1; ++i) { cnot_g(s_re, s_im, i, i + 1, tid); __syncthreads(); }
        cnot_g(s_re, s_im, NQ - 1, 0, tid);
        __syncthreads();

        // <Z_q> readout
        if (tid < NQ) s_acc[tid] = 0.f;
        __syncthreads();

        float z[NQ];
#pragma unroll
        for (int q = 0; q < NQ; ++q) z[q] = 0.f;
#pragma unroll
        for (int it = 0; it < QDIM / 256; ++it) {
            const int i = tid + it * 256;
            const float p = s_re[i] * s_re[i] + s_im[i] * s_im[i];
#pragma unroll
            for (int q = 0; q < NQ; ++q)
                z[q] += ((i >> (NQ - 1 - q)) & 1) ? -p : p;
        }
        // wave32 reduction; then lanes 0..11 of each wave add one qubit each
#pragma unroll
        for (int q = 0; q < NQ; ++q) {
#pragma unroll
            for (int off = 16; off > 0; off >>= 1)
                z[q] += __shfl_xor(z[q], off, 32);
        }
        const int wl = tid & 31;
        if (wl < NQ) atomicAdd(&s_acc[wl], z[wl]);
        __syncthreads();

        if (tid < NQ) ro[(k * BATCH + b) * NQ + tid] = s_acc[tid];
        __syncthreads();
    }
}

// ---------------------------------------------------------------------------
// Kernel 3: head GEMM via WMMA: [1536 x 12] x [12 x 8(pad16)]
//   grid = 12 blocks x 8 waves, one 16-row tile per wave (96 tiles)
//   K runs over {0,4,8} only: the 12..15 padding step is identically zero.
// ---------------------------------------------------------------------------
__global__ __launch_bounds__(256) void head_kernel(
    const float* __restrict__ ro,   // [1536][12]
    const float* __restrict__ hw,   // [8][12]
    const float* __restrict__ hb,   // [8]
    float* __restrict__ logits,     // [1536][8]  (== step_logits [6][256][8])
    float* __restrict__ fin)        // [256][8]
{
    __shared__ float hw_s[16][NQ];  // rows 8..15 = 0
    const int tid = threadIdx.x;
    if (tid < 16 * NQ) {
        const int r = tid / NQ, cc = tid % NQ;
        hw_s[r][cc] = (r < NCLS) ? hw[r * NQ + cc] : 0.f;
    }
    __syncthreads();

    const int tile = blockIdx.x * (blockDim.x >> 5) + (tid >> 5);
    const int lane = tid & 31;
    const int half = lane >> 4;
    const int l16  = lane & 15;

    v8f c = {0.f, 0.f, 0.f, 0.f, 0.f, 0.f, 0.f, 0.f};
    const int rowA = 16 * tile + l16;

#pragma unroll
    for (int k0 = 0; k0 < NQ; k0 += 4) {
        const int ka = k0 + 2 * half;                       // 0..10, even
        const v2f a  = *(const v2f*)&ro[rowA * NQ + ka];    // 8B aligned
        const v2f bm = *(const v2f*)&hw_s[l16][ka];         // ds_load_b64
        c = __builtin_amdgcn_wmma_f32_16x16x4_f32(
                false, a, false, bm, (short)0, c, false, false);
    }

    if (l16 < NCLS) {
        const float bias = hb[l16];
#pragma unroll
        for (int j = 0; j < 8; ++j) {
            const int row = 16 * tile + j + 8 * half;   // row = k*256 + b
            const float v = c[j] + bias;
            logits[row * NCLS + l16] = v;
            const int kk = row >> 8;
            const int bb = row & 255;
            if (kk == KSTEPS - 1) fin[bb * NCLS + l16] = v;
        }
    }
}

// ---------------------------------------------------------------------------
extern "C" void kernel_launch(void* const* d_in, const int* in_sizes, int n_in,
                              void* d_out, int out_size, void* d_ws, size_t ws_size,
                              hipStream_t stream) {
    const int*   ids   = (const int*)  d_in[0];
    const int*   amask = (const int*)  d_in[1];
    const float* emb   = (const float*)d_in[2];
    const float* pw    = (const float*)d_in[3];
    const float* pb    = (const float*)d_in[4];
    const float* theta = (const float*)d_in[5];
    const float* hw    = (const float*)d_in[6];
    const float* hb    = (const float*)d_in[7];

    float* out  = (float*)d_out;
    float* x_ws = (float*)d_ws;                         // 256*12 floats (12 KB)

    float* ro  = out + KSTEPS * BATCH * NCLS;           // +12288: readouts [6][256][12]
    float* fin = ro  + KSTEPS * BATCH * NQ;             // +30720: final logits [256][8]

    encode_kernel<<<BATCH / 16, 256, 0, stream>>>(ids, amask, emb, pw, pb, x_ws);
    qsim_kernel<<<BATCH, 256, 0, stream>>>(x_ws, theta, ro);
    head_kernel<<<(KSTEPS * BATCH / 16) / 8, 256, 0, stream>>>(ro, hw, hb, out, fin);
}